// Conv2d_symmetric_66314295050377
// MI455X (gfx1250) — compile-verified
//
#include <hip/hip_runtime.h>
#include <hip/hip_bf16.h>

// Depthwise symmetric 7x7 conv on (16,128,224,224) fp32, weight shared across
// channels -> 2048 independent single-channel 224x224 convolutions.
//
// Strategy: express each 16x16 output tile as a sum of Toeplitz GEMMs and run
// them on V_WMMA_F32_16X16X4_F32:
//   D[m,n] = sum_i sum_u A_i[m,u] * B_i[u,n]
//   A_i[m,u] = K(i, u-m)  (banded 16x24 weight slab, per kernel row i)
//   B_i[u,n] = x(h0-3 + th*16 + n + i, c0-3 + tw*16 + u)
// 6 K=4 chunks per kernel row x 7 rows = 42 WMMAs per 16x16 tile.

#define KS   7
#define PAD  3
#define H    224
#define W    224
#define TILE_R 32            // output rows per block
#define TILE_C 112           // output cols per block
#define LROWS  (TILE_R + 6)  // 38 halo rows
#define LCOLS  (TILE_C + 8)  // 120 cols loaded (6 halo + 2 zero-band cols)
#define LDSW   121           // LDS row stride (odd -> conflict-free gathers)
#define S2     113           // LDS stride for output transpose staging
#define NTHREADS 448         // 14 waves: 7 col-tiles x 2 row-tiles

typedef float v2f __attribute__((ext_vector_type(2)));
typedef float v8f __attribute__((ext_vector_type(8)));

__global__ __launch_bounds__(NTHREADS)
void dwconv7_sym_wmma(const float* __restrict__ x,
                      const float* __restrict__ kv,
                      float* __restrict__ out) {
    __shared__ float tile[LROWS * LDSW];   // reused for output staging
    __shared__ float kw[KS * KS];

    const int tid = threadIdx.x;
    const int c0  = blockIdx.x * TILE_C;
    const int h0  = blockIdx.y * TILE_R;
    const size_t img = blockIdx.z;
    const float* __restrict__ xin = x + img * (size_t)(H * W);
    float* __restrict__ o = out + img * (size_t)(H * W);

    // Build full symmetric 7x7 weight table: K[f] = kv[min(f, 48-f)]
    if (tid < KS * KS) {
        int f = tid;
        int c = (f <= 24) ? f : (48 - f);
        kw[f] = kv[c];
    }

    // Cooperative zero-padded halo load: rows [h0-3, h0+34], cols [c0-3, c0+116]
    for (int e = tid; e < LROWS * LCOLS; e += NTHREADS) {
        int r  = e / LCOLS;
        int cc = e - r * LCOLS;
        int gr = h0 - PAD + r;
        int gc = c0 - PAD + cc;
        float v = 0.0f;
        if (gr >= 0 && gr < H && gc >= 0 && gc < W) v = xin[gr * W + gc];
        tile[r * LDSW + cc] = v;
    }
    __syncthreads();

    const int lane = tid & 31;
    const int wv   = tid >> 5;          // 0..13
    const int th   = (wv >= 7) ? 1 : 0; // row tile within strip
    const int tw   = wv - 7 * th;       // col tile within strip
    const int lo   = lane & 15;
    const int hi   = lane >> 4;

    // Toeplitz A fragments (weights), built once per wave in registers.
    // fp32 16x4 A layout: lane = M (lo), VGPR v holds K = v + 2*hi.
    v2f A[KS][6];
#pragma unroll
    for (int i = 0; i < KS; ++i) {
#pragma unroll
        for (int uc = 0; uc < 6; ++uc) {
#pragma unroll
            for (int vv = 0; vv < 2; ++vv) {
                int u = 4 * uc + vv + 2 * hi;   // column index 0..23
                int j = u - lo;                 // kernel tap
                float a = (j >= 0 && j < KS) ? kw[i * KS + j] : 0.0f;
                A[i][uc][vv] = a;
            }
        }
    }

    // Accumulate 42 WMMAs. B layout mirrors A: lane = N (lo), VGPR v -> K = v + 2*hi.
    v8f C = {};
    const int brow0 = th * 16 + lo;        // + i  -> LDS row
    const int bcol0 = tw * 16 + 2 * hi;    // + 4*uc + vv -> LDS col
#pragma unroll
    for (int i = 0; i < KS; ++i) {
        const float* bp = &tile[(brow0 + i) * LDSW + bcol0];
#pragma unroll
        for (int uc = 0; uc < 6; ++uc) {
            v2f B;
            B[0] = bp[4 * uc + 0];
            B[1] = bp[4 * uc + 1];
            C = __builtin_amdgcn_wmma_f32_16x16x4_f32(
                    /*neg_a=*/false, A[i][uc],
                    /*neg_b=*/false, B,
                    /*c_mod=*/(short)0, C,
                    /*reuse_a=*/false, /*reuse_b=*/false);
        }
    }

    __syncthreads();  // all waves done reading input tile -> reuse LDS

    // Transpose D through LDS for coalesced stores.
    // C/D layout: VGPR r, lane -> D[M = r + 8*hi, N = lo]; M = out col, N = out row.
#pragma unroll
    for (int r = 0; r < 8; ++r) {
        int m = r + 8 * hi;
        tile[(th * 16 + lo) * S2 + tw * 16 + m] = C[r];
    }
    __syncthreads();

    // Coalesced global store: 32 rows x 112 contiguous floats.
    for (int e = tid; e < TILE_R * TILE_C; e += NTHREADS) {
        int r  = e / TILE_C;
        int cc = e - r * TILE_C;
        o[(size_t)(h0 + r) * W + (c0 + cc)] = tile[r * S2 + cc];
    }
}

extern "C" void kernel_launch(void* const* d_in, const int* in_sizes, int n_in,
                              void* d_out, int out_size, void* d_ws, size_t ws_size,
                              hipStream_t stream) {
    const float* x  = (const float*)d_in[0];
    const float* kv = (const float*)d_in[1];
    float* out = (float*)d_out;
    int nimg = in_sizes[0] / (H * W);     // 16*128 = 2048 channel-images
    dim3 grid(W / TILE_C, H / TILE_R, nimg);
    dim3 block(NTHREADS);
    hipLaunchKernelGGL(dwconv7_sym_wmma, grid, block, 0, stream, x, kv, out);
}